// SimpleMPNN_66546223284479
// MI455X (gfx1250) — compile-verified
//
#include <hip/hip_runtime.h>
#include <hip/hip_bf16.h>
#include <math.h>

// ---------------------------------------------------------------------------
// SimpleMPNN on MI455X (gfx1250): fp32 WMMA (16x16x4) GEMMs + sparse
// edge aggregation. All GEMMs use V_WMMA_F32_16X16X4_F32 — the problem is
// latency-bound (≈90 MFLOP, ≈8MB/step streamed, L2-resident), so fp32 WMMA
// gives exact accuracy at no bandwidth cost.
// ---------------------------------------------------------------------------

typedef float v2f __attribute__((ext_vector_type(2)));
typedef float v8f __attribute__((ext_vector_type(8)));

#define N_NODES 1024
#define HID 64
#define IN_DIM 16
#define STEPS 3

// C[M x N] = act( A[M x K] (row-major, lda) * B[N x K]^T (row-major, ldb) + bias )
// One wave (32 threads) per 16x16 output tile. grid = (M/16, N/16).
// WMMA f32 16x16x4 VGPR layout (CDNA5 ISA 7.12.2):
//   A: lane l, comp j -> A[l%16][2*(l/16)+j]
//   B: lane l, comp j -> B_col[l%16] at K=2*(l/16)+j   (B given as N x K row-major)
//   C: vgpr r, lane l -> C[r + 8*(l/16)][l%16]
__global__ __launch_bounds__(32)
void wmma_gemm_bias_act(const float* __restrict__ A,
                        const float* __restrict__ B,
                        const float* __restrict__ bias,
                        float* __restrict__ C,
                        int K, int lda, int ldb, int ldc, int act)
{
    const int lane = threadIdx.x;           // 0..31, wave32
    const int tm   = blockIdx.x << 4;       // tile row base (M)
    const int tn   = blockIdx.y << 4;       // tile col base (N)
    const int r16  = lane & 15;
    const int kh   = (lane >> 4) << 1;      // 0 or 2

    const float* __restrict__ a0 = A + (size_t)(tm + r16) * lda + kh;
    const float* __restrict__ b0 = B + (size_t)(tn + r16) * ldb + kh;

    v8f acc = {};
    for (int kk = 0; kk < K; kk += 4) {
        v2f a, b;
        a.x = a0[kk + 0]; a.y = a0[kk + 1];
        b.x = b0[kk + 0]; b.y = b0[kk + 1];
        acc = __builtin_amdgcn_wmma_f32_16x16x4_f32(
            /*neg_a=*/false, a, /*neg_b=*/false, b,
            /*c_mod=*/(short)0, acc, /*reuse_a=*/false, /*reuse_b=*/false);
    }

    const int mbase = tm + ((lane >> 4) << 3);
    const int n     = tn + r16;
    const float bv  = bias[n];
#pragma unroll
    for (int r = 0; r < 8; ++r) {
        float v = acc[r] + bv;
        if (act == 1) v = tanhf(v);
        C[(size_t)(mbase + r) * ldc + n] = v;
    }
}

// msgs[u] = (1/deg(u)) * sum_{v: A[u,v]>0} tanh(proj[v] + E[u,v]*w_e), else 0.
// One block per node u, 64 threads (one per hidden dim). A/E rows staged in LDS;
// branch on A[u,v] is uniform across the block.
__global__ __launch_bounds__(64)
void aggregate(const float* __restrict__ A,
               const float* __restrict__ E,
               const float* __restrict__ proj,
               const float* __restrict__ W_msg,   // HID x (HID+1); last col = w_e
               float* __restrict__ msgs)
{
    __shared__ float sA[N_NODES];
    __shared__ float sE[N_NODES];
    const int u = blockIdx.x;
    const int j = threadIdx.x;

    const float* __restrict__ Arow = A + (size_t)u * N_NODES;
    const float* __restrict__ Erow = E + (size_t)u * N_NODES;
    for (int v = j; v < N_NODES; v += 64) {
        sA[v] = Arow[v];
        sE[v] = Erow[v];
    }
    __syncthreads();

    const float we = W_msg[(size_t)j * (HID + 1) + HID];
    float acc = 0.0f;
    int   deg = 0;
    for (int v = 0; v < N_NODES; ++v) {
        if (sA[v] > 0.0f) {
            acc += tanhf(proj[(size_t)v * HID + j] + sE[v] * we);
            ++deg;
        }
    }
    msgs[(size_t)u * HID + j] = (deg > 0) ? acc / (float)deg : 0.0f;
}

__device__ __forceinline__ float sigmoidf_(float x) {
    return 1.0f / (1.0f + __expf(-x));
}

// GRU gates: gi = msgs@W_ih^T + b_ih (precomputed, N x 192),
//            gh = h   @W_hh^T + b_hh (precomputed, N x 192)
__global__ __launch_bounds__(256)
void gru_update(const float* __restrict__ gi,
                const float* __restrict__ gh,
                const float* __restrict__ h,
                float* __restrict__ hout)
{
    const int idx = blockIdx.x * 256 + threadIdx.x;   // 0 .. N*HID-1
    const int n = idx >> 6;
    const int j = idx & 63;
    const float* __restrict__ gin = gi + (size_t)n * (3 * HID);
    const float* __restrict__ ghn = gh + (size_t)n * (3 * HID);

    const float r   = sigmoidf_(gin[j]           + ghn[j]);
    const float z   = sigmoidf_(gin[HID + j]     + ghn[HID + j]);
    const float nct = tanhf    (gin[2 * HID + j] + r * ghn[2 * HID + j]);
    const float hp  = h[idx];
    hout[idx] = (1.0f - z) * nct + z * hp;
}

// Readout: cat = [h[src]; h[tgt]] (128); hid = relu(W_r1@cat + b_r1) (64);
// out = sigmoid(dot(W_r2, hid) + b_r2). One block of 128 threads.
__global__ __launch_bounds__(128)
void readout(const float* __restrict__ h,
             const float* __restrict__ W_r1, const float* __restrict__ b_r1,
             const float* __restrict__ W_r2, const float* __restrict__ b_r2,
             const int* __restrict__ src, const int* __restrict__ tgt,
             float* __restrict__ out)
{
    __shared__ float cat[2 * HID];
    __shared__ float hid[HID];
    const int t = threadIdx.x;
    const int s = src[0];
    const int g = tgt[0];
    cat[t] = (t < HID) ? h[(size_t)s * HID + t] : h[(size_t)g * HID + (t - HID)];
    __syncthreads();
    if (t < HID) {
        float acc = b_r1[t];
        const float* __restrict__ wr = W_r1 + (size_t)t * (2 * HID);
#pragma unroll 8
        for (int k = 0; k < 2 * HID; ++k) acc += wr[k] * cat[k];
        hid[t] = fmaxf(acc, 0.0f);
    }
    __syncthreads();
    if (t == 0) {
        float acc = b_r2[0];
#pragma unroll 8
        for (int k = 0; k < HID; ++k) acc += W_r2[k] * hid[k];
        out[0] = 1.0f / (1.0f + __expf(-acc));
    }
}

extern "C" void kernel_launch(void* const* d_in, const int* in_sizes, int n_in,
                              void* d_out, int out_size, void* d_ws, size_t ws_size,
                              hipStream_t stream)
{
    const float* A     = (const float*)d_in[0];
    const float* E     = (const float*)d_in[1];
    const float* X     = (const float*)d_in[2];
    const float* W_enc = (const float*)d_in[3];
    const float* b_enc = (const float*)d_in[4];
    const float* W_msg = (const float*)d_in[5];
    const float* b_msg = (const float*)d_in[6];
    const float* W_ih  = (const float*)d_in[7];
    const float* W_hh  = (const float*)d_in[8];
    const float* b_ih  = (const float*)d_in[9];
    const float* b_hh  = (const float*)d_in[10];
    const float* W_r1  = (const float*)d_in[11];
    const float* b_r1  = (const float*)d_in[12];
    const float* W_r2  = (const float*)d_in[13];
    const float* b_r2  = (const float*)d_in[14];
    const int*   src   = (const int*)d_in[15];
    const int*   tgt   = (const int*)d_in[16];
    (void)in_sizes; (void)n_in; (void)out_size; (void)ws_size;

    // Workspace layout (floats)
    float* ws   = (float*)d_ws;
    float* hA   = ws;                          // N*HID
    float* hB   = hA + N_NODES * HID;          // N*HID
    float* proj = hB + N_NODES * HID;          // N*HID
    float* msgs = proj + N_NODES * HID;        // N*HID
    float* gi   = msgs + N_NODES * HID;        // N*3*HID
    float* gh   = gi + N_NODES * 3 * HID;      // N*3*HID

    const dim3 wv(32);

    // Encoder: h = tanh(X @ W_enc^T + b_enc); M=1024, N=64, K=16
    wmma_gemm_bias_act<<<dim3(N_NODES / 16, HID / 16), wv, 0, stream>>>(
        X, W_enc, b_enc, hA, /*K=*/IN_DIM, /*lda=*/IN_DIM, /*ldb=*/IN_DIM,
        /*ldc=*/HID, /*act=*/1);

    float* cur = hA;
    float* nxt = hB;
    for (int step = 0; step < STEPS; ++step) {
        // proj = cur @ Wm_h^T + b_msg  (Wm_h = W_msg[:, :HID], ldb = HID+1)
        wmma_gemm_bias_act<<<dim3(N_NODES / 16, HID / 16), wv, 0, stream>>>(
            cur, W_msg, b_msg, proj, /*K=*/HID, /*lda=*/HID, /*ldb=*/HID + 1,
            /*ldc=*/HID, /*act=*/0);

        // sparse mean-aggregation of tanh(proj[v] + E[u,v]*w_e)
        aggregate<<<dim3(N_NODES), dim3(64), 0, stream>>>(A, E, proj, W_msg, msgs);

        // gi = msgs @ W_ih^T + b_ih ; gh = cur @ W_hh^T + b_hh  (N=192, K=64)
        wmma_gemm_bias_act<<<dim3(N_NODES / 16, (3 * HID) / 16), wv, 0, stream>>>(
            msgs, W_ih, b_ih, gi, /*K=*/HID, /*lda=*/HID, /*ldb=*/HID,
            /*ldc=*/3 * HID, /*act=*/0);
        wmma_gemm_bias_act<<<dim3(N_NODES / 16, (3 * HID) / 16), wv, 0, stream>>>(
            cur, W_hh, b_hh, gh, /*K=*/HID, /*lda=*/HID, /*ldb=*/HID,
            /*ldc=*/3 * HID, /*act=*/0);

        // h' = (1-z)*tanh(in + r*hn) + z*h
        gru_update<<<dim3((N_NODES * HID) / 256), dim3(256), 0, stream>>>(
            gi, gh, cur, nxt);

        float* tmp = cur; cur = nxt; nxt = tmp;
    }

    readout<<<dim3(1), dim3(128), 0, stream>>>(cur, W_r1, b_r1, W_r2, b_r2,
                                               src, tgt, (float*)d_out);
}